// Hand_Input_Sorter_48636209660396
// MI455X (gfx1250) — compile-verified
//
#include <hip/hip_runtime.h>
#include <stdint.h>

// CSV_Architecture_v2 frame: 130 floats = two 65-float hand blocks.
#define FRAME      130
#define HALF       65
#define FPB        16                   // frames per block (tile)
#define TILE_FLOATS (FRAME * FPB)       // 2080 floats
#define TILE_CHUNKS (TILE_FLOATS / 4)   // 520 x 16B chunks (8320 B, 16B-aligned tiles)
#define NTHREADS   256                  // 8 waves (wave32)

typedef float v4f __attribute__((ext_vector_type(4)));
typedef int   v4i __attribute__((ext_vector_type(4)));

// Pointer types matching the async-LDS builtin signature (per round-1 diagnostic:
// param is int4* in the global AS; LDS side is the AS(3) analog).
typedef __attribute__((address_space(1))) v4i* gptr_v4i;
typedef __attribute__((address_space(3))) v4i* lptr_v4i;

// ---- CDNA5 async global->LDS path (probe-guarded so compile never breaks) ----
#if defined(__gfx1250__) && __has_builtin(__builtin_amdgcn_global_load_async_to_lds_b128)
#define USE_ASYNC_LDS 1
#else
#define USE_ASYNC_LDS 0
#endif

__device__ __forceinline__ void wait_asynccnt0() {
#if defined(__gfx1250__)
#if __has_builtin(__builtin_amdgcn_s_wait_asynccnt)
    __builtin_amdgcn_s_wait_asynccnt(0);
#else
    asm volatile("s_wait_asynccnt 0" ::: "memory");
#endif
#endif
}

// skip predicate (JAX/IEEE semantics: NaN compares false; h!=h detects NaN)
__device__ __forceinline__ int compute_skip(float h0, float p0, float h1, float p1) {
    const bool nan0 = (h0 != h0);
    const bool nan1 = (h1 != h1);
    const bool no_nan = !nan0 && !nan1;
    const bool eq = (h0 == h1);
    const bool skip =
        (h1 > h0) ||                                    // skip1: already ordered
        (nan0 && nan1) ||                               // skip2: no hands
        (nan0 && (h1 == 1.0f)) ||                       // skip3: only hand1, right
        (nan1 && (h0 == 0.0f)) ||                       // skip4: only hand0, left
        (no_nan && eq && (h0 == 0.0f) && (p0 > p1)) ||  // skip5: both left, p0 wins
        (no_nan && eq && (h0 == 1.0f) && (p0 < p1));    // skip6: both right, p1 wins
    return skip ? 1 : 0;
}

__global__ void __launch_bounds__(NTHREADS)
hand_input_sorter_kernel(const float* __restrict__ in,
                         float* __restrict__ out,
                         long long nFloats) {
    __shared__ __align__(16) float lds[TILE_FLOATS];
    __shared__ int skipArr[FPB];

    const int tid = threadIdx.x;
    const long long base = (long long)blockIdx.x * TILE_FLOATS;
    const long long remain = nFloats - base;
    const int tileFloats = (remain >= (long long)TILE_FLOATS) ? TILE_FLOATS : (int)remain;
    const bool fullTile = (tileFloats == TILE_FLOATS);

    // ---- Phase 1: stage tile into LDS (linear, 16B-vectorized global reads) ----
    if (fullTile) {
#if USE_ASYNC_LDS
        for (int c = tid; c < TILE_CHUNKS; c += NTHREADS) {
            __builtin_amdgcn_global_load_async_to_lds_b128(
                (gptr_v4i)(uintptr_t)(in + base + 4 * c),
                (lptr_v4i)(uint32_t)(uintptr_t)&lds[4 * c],
                /*offset=*/0, /*cpol=*/0);
        }
        wait_asynccnt0();   // this wave's ASYNCcnt -> 0, then barrier covers all waves
#else
        for (int c = tid; c < TILE_CHUNKS; c += NTHREADS) {
            *(v4f*)&lds[4 * c] = *(const v4f*)(in + base + 4 * c);
        }
#endif
    } else {
        for (int i = tid; i < tileFloats; i += NTHREADS) {
            lds[i] = in[base + i];
        }
    }
    __syncthreads();

    // ---- Phase 2: one flag per frame ----
    const int framesInTile = tileFloats / FRAME;
    if (tid < framesInTile) {
        const int fb = tid * FRAME;
        skipArr[tid] = compute_skip(lds[fb + 0], lds[fb + 1],
                                    lds[fb + HALF], lds[fb + HALF + 1]);
    }
    __syncthreads();

    // ---- Phase 3: permuted gather from LDS, aligned non-temporal b128 stores ----
    if (fullTile) {
        for (int c = tid; c < TILE_CHUNKS; c += NTHREADS) {
            const int e = 4 * c;
            v4f v;
#pragma unroll
            for (int k = 0; k < 4; ++k) {
                const int ee = e + k;
                const int f = ee / FRAME;            // constant div -> mulhi
                const int i = ee - f * FRAME;
                const int src = skipArr[f] ? i : (i >= HALF ? i - HALF : i + HALF);
                v[k] = lds[f * FRAME + src];
            }
            __builtin_nontemporal_store(v, (v4f*)(out + base + e));
        }
    } else {
        for (int ee = tid; ee < tileFloats; ee += NTHREADS) {
            const int f = ee / FRAME;
            const int i = ee - f * FRAME;
            const int src = skipArr[f] ? i : (i >= HALF ? i - HALF : i + HALF);
            out[base + ee] = lds[f * FRAME + src];
        }
    }
}

extern "C" void kernel_launch(void* const* d_in, const int* in_sizes, int n_in,
                              void* d_out, int out_size, void* d_ws, size_t ws_size,
                              hipStream_t stream) {
    (void)n_in; (void)out_size; (void)d_ws; (void)ws_size;
    const float* in = (const float*)d_in[0];   // X, float32 [B,S,130] flat
    float* out = (float*)d_out;                // float32, same shape
    const long long nFloats = (long long)in_sizes[0];
    const long long frames = nFloats / FRAME;
    const int blocks = (int)((frames + FPB - 1) / FPB);
    hand_input_sorter_kernel<<<blocks, NTHREADS, 0, stream>>>(in, out, nFloats);
}